// AutoregressiveDecoder_26886495272996
// MI455X (gfx1250) — compile-verified
//
#include <hip/hip_runtime.h>
#include <cstdint>

// Problem constants (from the reference)
#define B_   2048
#define T_   1024
#define D_   32
#define S_   2
// Blocked-scan decomposition
#define K_   32               // time steps per chunk
#define NCH  (T_ / K_)        // 32 chunks per batch row
#define WG1  128              // phase-1 workgroup (4 waves)
#define G_   2                // z rows staged per async stage
#define NSTG (K_ / G_)        // 16 stages per chunk

// ---------------------------------------------------------------------------
// Phase 1: for each (b, chunk) task, run the recurrence from a ZERO state,
// storing local prefixes p_j into out[b, c*K + j].  z is staged into LDS with
// double-buffered gfx1250 async loads (ASYNCcnt), then consumed via
// ds_load_b128.  No cross-lane sharing -> no barriers, only s_wait_asynccnt.
// z is a 256MB single-use stream (> 192MB L2), so the async loads are issued
// non-temporal to keep L2 reserved for the out[] working set that phases 2/3
// re-read.
// ---------------------------------------------------------------------------
__global__ __launch_bounds__(WG1) void ar_phase1(
    const float* __restrict__ z, const float* __restrict__ W,
    float* __restrict__ out)
{
  // [2 buffers][G rows][8 q-slots][WG1 lanes] of float4  = 64 KB
  __shared__ float4 smem[2 * G_ * 8 * WG1];

  const int tid  = threadIdx.x;
  const int task = blockIdx.x * WG1 + tid;
  const int b    = task & (B_ - 1);   // consecutive b within a wave
  const int c    = task >> 11;        // task / B_ : chunk id (uniform per wave)

  // Weights are wave-uniform -> scalar loads (confirmed: s_load_b512 x4).
  // W layout: (2, 34) row-major.  Ws = W[:, :2], Wz = W[:, 2:].
  const float ws00 = W[0],  ws01 = W[1];
  const float ws10 = W[34], ws11 = W[35];
  float wz0[D_], wz1[D_];
#pragma unroll
  for (int d = 0; d < D_; ++d) { wz0[d] = W[2 + d]; wz1[d] = W[36 + d]; }

  // Per-lane base addresses
  const uint64_t zbase =
      (uint64_t)(uintptr_t)(z + ((size_t)b * T_ + (size_t)c * K_) * D_);
  const uint32_t lds_lane =
      (uint32_t)(uintptr_t)(&smem[0]) + (uint32_t)tid * 16u;

  // Issue one stage: G rows x 8 x b128 non-temporal async copies into
  // buffer (s & 1).
  auto issue = [&](int s) {
    const int buf = s & 1;
    const uint64_t g0 = zbase + (uint64_t)s * (G_ * D_ * 4);
#pragma unroll
    for (int r = 0; r < G_; ++r) {
#pragma unroll
      for (int q = 0; q < 8; ++q) {
        uint32_t lds = lds_lane +
            (uint32_t)((((buf * G_ + r) * 8 + q) * WG1) * 16);
        uint64_t ga = g0 + (uint64_t)(r * (D_ * 4) + q * 16);
        asm volatile("global_load_async_to_lds_b128 %0, %1, off th:TH_LOAD_NT"
                     :: "v"(lds), "v"(ga) : "memory");
      }
    }
  };

  issue(0);
  issue(1);

  const int Kc = (c == NCH - 1) ? (K_ - 1) : K_;   // last chunk: t stops at 1023
  float2* __restrict__ op = (float2*)out + ((size_t)b * T_ + (size_t)c * K_);

  float p0 = 0.f, p1 = 0.f;   // local prefix state (zero initial state)
  for (int s = 0; s < NSTG; ++s) {
    // Wait for stage s to land in LDS (16 async ops may still be in flight
    // for stage s+1).
    if (s + 1 < NSTG) asm volatile("s_wait_asynccnt 16" ::: "memory");
    else              asm volatile("s_wait_asynccnt 0"  ::: "memory");

    const int buf = s & 1;
#pragma unroll
    for (int r = 0; r < G_; ++r) {
      const int base = ((buf * G_ + r) * 8) * WG1 + tid;
      float a0[2] = {0.f, 0.f};   // partial accumulators for u0
      float a1[2] = {0.f, 0.f};   // partial accumulators for u1
#pragma unroll
      for (int q = 0; q < 8; ++q) {
        float4 v = smem[base + q * WG1];   // ds_load_b128
        const int d = q * 4;
        const int h = q >> 2;
        a0[h] = fmaf(wz0[d + 0], v.x, a0[h]);
        a0[h] = fmaf(wz0[d + 1], v.y, a0[h]);
        a0[h] = fmaf(wz0[d + 2], v.z, a0[h]);
        a0[h] = fmaf(wz0[d + 3], v.w, a0[h]);
        a1[h] = fmaf(wz1[d + 0], v.x, a1[h]);
        a1[h] = fmaf(wz1[d + 1], v.y, a1[h]);
        a1[h] = fmaf(wz1[d + 2], v.z, a1[h]);
        a1[h] = fmaf(wz1[d + 3], v.w, a1[h]);
      }
      const float u0 = a0[0] + a0[1];
      const float u1 = a1[0] + a1[1];
      const float np0 = fmaf(ws00, p0, fmaf(ws01, p1, u0));
      const float np1 = fmaf(ws10, p0, fmaf(ws11, p1, u1));
      p0 = np0; p1 = np1;
      const int j = s * G_ + r + 1;      // 1-based step within chunk
      if (j <= Kc) op[j] = make_float2(p0, p1);   // global_store_b64
    }

    // Ensure our ds_loads of this buffer are fully done before the async
    // engine overwrites it with stage s+2.
    asm volatile("s_wait_dscnt 0" ::: "memory");
    if (s + 2 < NSTG) issue(s + 2);
  }
}

// ---------------------------------------------------------------------------
// Phase 2: per batch row, sequentially combine the 32 chunk boundaries.
// boundary[c] = s_{c*K};  s_{(c+1)K} = Ws^K * s_{cK} + p_K(chunk c),
// where p_K(chunk c) was stored by phase 1 at out[b, (c+1)*K].
// Also writes out[b,0] = init.
// ---------------------------------------------------------------------------
__global__ __launch_bounds__(256) void ar_phase2(
    const float* __restrict__ init, const float* __restrict__ W,
    float* __restrict__ out, float2* __restrict__ bnd)
{
  const int b = blockIdx.x * 256 + threadIdx.x;
  if (b >= B_) return;

  const float ws00 = W[0],  ws01 = W[1];
  const float ws10 = W[34], ws11 = W[35];
  // M = Ws^K  (K = 32 = 2^5 -> 5 squarings of a 2x2)
  float m00 = ws00, m01 = ws01, m10 = ws10, m11 = ws11;
#pragma unroll
  for (int i = 0; i < 5; ++i) {
    const float t00 = fmaf(m00, m00, m01 * m10);
    const float t01 = fmaf(m00, m01, m01 * m11);
    const float t10 = fmaf(m10, m00, m11 * m10);
    const float t11 = fmaf(m10, m01, m11 * m11);
    m00 = t00; m01 = t01; m10 = t10; m11 = t11;
  }

  const float2* __restrict__ po = (const float2*)out + (size_t)b * T_;
  float2 s = ((const float2*)init)[b];
  bnd[b * NCH + 0] = s;
  ((float2*)out)[(size_t)b * T_] = s;    // out[b, 0] = init_states[b]
#pragma unroll
  for (int c = 1; c < NCH; ++c) {
    const float2 pk = po[c * K_];        // p_K of chunk c-1 (written by phase1)
    float2 ns;
    ns.x = fmaf(m00, s.x, fmaf(m01, s.y, pk.x));
    ns.y = fmaf(m10, s.x, fmaf(m11, s.y, pk.y));
    s = ns;
    bnd[b * NCH + c] = s;
  }
}

// ---------------------------------------------------------------------------
// Phase 3: per (b, chunk), add Ws^j * boundary to the stored local prefixes:
// out[b, c*K + j] += Ws^j * s_{c*K}   for j = 1..Kc.
// The out[] lines were written by phase 1 and (with z streamed NT) should
// still be L2-resident, so this RMW runs at L2 bandwidth.
// ---------------------------------------------------------------------------
__global__ __launch_bounds__(256) void ar_phase3(
    const float* __restrict__ W, const float2* __restrict__ bnd,
    float* __restrict__ out)
{
  const int task = blockIdx.x * 256 + threadIdx.x;
  const int b    = task & (B_ - 1);
  const int c    = task >> 11;

  const float ws00 = W[0],  ws01 = W[1];
  const float ws10 = W[34], ws11 = W[35];

  float2 v = bnd[b * NCH + c];
  const int Kc = (c == NCH - 1) ? (K_ - 1) : K_;
  float2* __restrict__ op = (float2*)out + ((size_t)b * T_ + (size_t)c * K_);
#pragma unroll
  for (int j = 1; j <= K_; ++j) {
    const float vx = fmaf(ws00, v.x, ws01 * v.y);
    const float vy = fmaf(ws10, v.x, ws11 * v.y);
    v.x = vx; v.y = vy;
    if (j <= Kc) {
      float2 o = op[j];
      o.x += v.x; o.y += v.y;
      op[j] = o;
    }
  }
}

// ---------------------------------------------------------------------------
extern "C" void kernel_launch(void* const* d_in, const int* in_sizes, int n_in,
                              void* d_out, int out_size, void* d_ws, size_t ws_size,
                              hipStream_t stream) {
  (void)in_sizes; (void)n_in; (void)out_size; (void)ws_size;
  const float* init = (const float*)d_in[0];   // (B, S)
  const float* z    = (const float*)d_in[1];   // (B, T, D)
  const float* W    = (const float*)d_in[2];   // (S, S+D)
  float* out  = (float*)d_out;                 // (B, T, S)
  float2* bnd = (float2*)d_ws;                 // B * NCH boundaries (512 KB)

  ar_phase1<<<(B_ * NCH) / WG1, WG1, 0, stream>>>(z, W, out);
  ar_phase2<<<B_ / 256, 256, 0, stream>>>(init, W, out, bnd);
  ar_phase3<<<(B_ * NCH) / 256, 256, 0, stream>>>(W, bnd, out);
}